// Embedding_59682865545863
// MI455X (gfx1250) — compile-verified
//
#include <hip/hip_runtime.h>

// Embedding row-gather for MI455X (gfx1250, wave32).
//
// out[row, 0:128] = weight[idx[row], 0:128]   (f32, 512 bytes per row)
//
// One wave32 per row: lane L moves bytes [16L, 16L+16) of the row as a single
// b128 load + b128 nontemporal store. Index & source address are scalarized
// via readfirstlane (wave-uniform), next row is prefetched while the current
// row is in flight.

typedef float v4f __attribute__((ext_vector_type(4)));

#define WAVES_PER_BLOCK 8
#define BLOCK_THREADS (WAVES_PER_BLOCK * 32)
#define MAX_BLOCKS 8192
#define D_EMB 128  // floats per row

__global__ __launch_bounds__(BLOCK_THREADS) void embed_gather_kernel(
    const int* __restrict__ idx,
    const float* __restrict__ weight,
    float* __restrict__ out,
    int n_rows) {
  const int lane = threadIdx.x & 31;
  // Wave-uniform row id; force into SGPR so index load + address math scalarize.
  int row = __builtin_amdgcn_readfirstlane(
      (int)(blockIdx.x * WAVES_PER_BLOCK + (threadIdx.x >> 5)));
  const int stride = (int)(gridDim.x * WAVES_PER_BLOCK);

  if (row >= n_rows) return;

  int j = __builtin_amdgcn_readfirstlane(idx[row]);

  for (;;) {
    const int nrow = row + stride;
    const bool has_next = (nrow < n_rows);
    int jn = 0;
    if (has_next) {
      // Load next index early and prefetch that row's cachelines
      // (32 lanes x 16B stride spans the full 512B row).
      jn = __builtin_amdgcn_readfirstlane(idx[nrow]);
      __builtin_prefetch(weight + (size_t)jn * D_EMB + (size_t)lane * 4, 0, 3);
    }

    // Gather current row: one global_load_b128 per lane (SGPR base + lane offset).
    const v4f* __restrict__ src = (const v4f*)(weight + (size_t)j * D_EMB);
    v4f v = src[lane];

    // Streamed output: nontemporal store keeps the 256MB write stream out of L2.
    v4f* __restrict__ dst = (v4f*)(out + (size_t)row * D_EMB);
    __builtin_nontemporal_store(v, dst + lane);

    if (!has_next) break;
    row = nrow;
    j = jn;
  }
}

extern "C" void kernel_launch(void* const* d_in, const int* in_sizes, int n_in,
                              void* d_out, int out_size, void* d_ws, size_t ws_size,
                              hipStream_t stream) {
  const int* idx = (const int*)d_in[0];      // [N_IDX] int32
  const float* weight = (const float*)d_in[1];  // [N_EMB, 128] f32
  float* out = (float*)d_out;                // [N_IDX, 128] f32

  const int n_rows = in_sizes[0];

  int blocks = (n_rows + WAVES_PER_BLOCK - 1) / WAVES_PER_BLOCK;
  if (blocks > MAX_BLOCKS) blocks = MAX_BLOCKS;
  if (blocks < 1) blocks = 1;

  embed_gather_kernel<<<blocks, BLOCK_THREADS, 0, stream>>>(idx, weight, out, n_rows);
}